// LssEncoder_75453985456562
// MI455X (gfx1250) — compile-verified
//
#include <hip/hip_runtime.h>
#include <math.h>

typedef float v2f __attribute__((ext_vector_type(2)));
typedef float v8f __attribute__((ext_vector_type(8)));

// ---- problem constants (match reference) ----
static constexpr int   C_ = 128;
static constexpr int   H_ = 28;
static constexpr int   W_ = 50;
static constexpr int   D_ = 64;
static constexpr int   X_ = 64;
static constexpr int   Y_ = 64;
static constexpr int   Z_ = 64;
static constexpr int   N_ = 6;
static constexpr float DMIN_ = 1.0f;
static constexpr float BINF  = (float)(2.0 * (50.0 - 1.0) / (64.0 * 65.0));

// ------------------------------------------------------------------
// 3x3 inverse (adjugate)
// ------------------------------------------------------------------
__device__ __forceinline__ void inv3x3(const float* __restrict__ K, float* __restrict__ o) {
    const float a = K[0], b = K[1], c = K[2];
    const float d = K[3], e = K[4], f = K[5];
    const float g = K[6], h = K[7], i = K[8];
    const float A0 =  (e * i - f * h);
    const float A1 = -(d * i - f * g);
    const float A2 =  (d * h - e * g);
    const float r  = 1.0f / (a * A0 + b * A1 + c * A2);
    o[0] = A0 * r; o[1] = -(b * i - c * h) * r; o[2] =  (b * f - c * e) * r;
    o[3] = A1 * r; o[4] =  (a * i - c * g) * r; o[5] = -(a * f - c * d) * r;
    o[6] = A2 * r; o[7] = -(a * h - b * g) * r; o[8] =  (a * e - b * d) * r;
}

// ------------------------------------------------------------------
// Setup (1 wave): compute per-camera composed transform M_n = P_n @ invK4_n
// (3x4 each) into d_ws. The 6 tiny 4x4 matmuls are packed into two
// V_WMMA_F32_16X16X4_F32 ops (diagonal 4x4 blocks of the 16x16 product);
// a scalar recompute guards the result elementwise so correctness never
// depends on the (undocumented-for-f32) B-fragment layout.
// ------------------------------------------------------------------
__global__ __launch_bounds__(32) void lss_setup(const float* __restrict__ pose,
                                                const float* __restrict__ intr,
                                                float* __restrict__ M /* N_*12 */) {
    __shared__ float iK4[N_][16];
    __shared__ float P4 [N_][16];
    __shared__ float Msc[N_][12];
    const int lane = threadIdx.x;

    if (lane < N_) {
        float inv[9];
        inv3x3(intr + lane * 9, inv);
        const float m[16] = { inv[0], inv[1], inv[2], 0.f,
                              inv[3], inv[4], inv[5], 0.f,
                              inv[6], inv[7], inv[8], 0.f,
                              0.f,    0.f,    0.f,    1.f };
        #pragma unroll
        for (int j = 0; j < 16; ++j) {
            iK4[lane][j] = m[j];
            P4 [lane][j] = pose[lane * 16 + j];
        }
    }
    __syncthreads();

    if (lane < N_) {                       // scalar reference composition (3x4)
        #pragma unroll
        for (int i = 0; i < 3; ++i)
            #pragma unroll
            for (int j = 0; j < 4; ++j) {
                float s = 0.f;
                #pragma unroll
                for (int k = 0; k < 4; ++k) s += P4[lane][i * 4 + k] * iK4[lane][k * 4 + j];
                Msc[lane][i * 4 + j] = s;
            }
    }
    __syncthreads();

#if defined(__has_builtin) && __has_builtin(__builtin_amdgcn_wmma_f32_16x16x4_f32)
    __shared__ float Dfull[16][16];
    __shared__ float Dw[N_][12];
    const int half = lane >> 4;            // 0: lanes 0-15, 1: lanes 16-31
    const int l16  = lane & 15;

    for (int grp = 0; grp < 2; ++grp) {    // grp0: cams 0-3, grp1: cams 4-5 (+pad)
        // A 16x4 (ISA 7.12.2): lane=row; VGPR0/1 = K(0,1) for lanes 0-15, K(2,3) for 16-31
        v2f a, b;
        {
            const int camA = grp * 4 + (l16 >> 2);
            const int i    = l16 & 3;
            const int k0   = half * 2;
            const bool ok  = camA < N_;
            a.x = ok ? P4[camA][i * 4 + k0]     : 0.f;
            a.y = ok ? P4[camA][i * 4 + k0 + 1] : 0.f;
        }
        // B 4x16 (assumed mirror layout): lane=col; VGPR0/1 = K(0,1) / K(2,3) per lane half
        {
            const int camB = grp * 4 + (l16 >> 2);
            const int j    = l16 & 3;
            const int k0   = half * 2;
            const bool ok  = camB < N_;
            b.x = ok ? iK4[camB][k0 * 4 + j]       : 0.f;
            b.y = ok ? iK4[camB][(k0 + 1) * 4 + j] : 0.f;
        }
        v8f cacc = {};
        v8f dres = __builtin_amdgcn_wmma_f32_16x16x4_f32(false, a, false, b,
                                                         (short)0, cacc, false, false);
        // C/D 16x16 f32: VGPR v -> row v (lanes 0-15) / row v+8 (lanes 16-31), col = lane%16
        #pragma unroll
        for (int v = 0; v < 8; ++v) Dfull[v + 8 * half][l16] = dres[v];
        __syncthreads();

        for (int idx = lane; idx < 4 * 12; idx += 32) {
            const int cg  = idx / 12;
            const int cam = grp * 4 + cg;
            const int i   = (idx % 12) >> 2;
            const int j   = idx & 3;
            if (cam < N_) Dw[cam][i * 4 + j] = Dfull[cg * 4 + i][cg * 4 + j];
        }
        __syncthreads();
    }
    // Use the WMMA result where it matches the scalar path (ulp-level); else scalar.
    for (int idx = lane; idx < N_ * 12; idx += 32) {
        const int cam = idx / 12, e = idx % 12;
        const float sc = Msc[cam][e];
        const float wm = Dw[cam][e];
        M[idx] = (fabsf(wm - sc) <= 1e-4f * (1.0f + fabsf(sc))) ? wm : sc;
    }
#else
    for (int idx = lane; idx < N_ * 12; idx += 32) M[idx] = Msc[idx / 12][idx % 12];
#endif
}

// ------------------------------------------------------------------
// Zero-fill (global_store_b128)
// ------------------------------------------------------------------
__global__ __launch_bounds__(256) void lss_zero4(float4* __restrict__ out, int n4) {
    const int i = blockIdx.x * blockDim.x + threadIdx.x;
    if (i < n4) out[i] = make_float4(0.f, 0.f, 0.f, 0.f);
}
__global__ __launch_bounds__(256) void lss_zero1(float* __restrict__ out, int base, int n) {
    const int i = base + blockIdx.x * blockDim.x + threadIdx.x;
    if (i < n) out[i] = 0.f;
}

// ------------------------------------------------------------------
// Scatter: one thread per (pixel, channel); reads precomposed M from d_ws.
// ------------------------------------------------------------------
__global__ __launch_bounds__(256) void lss_scatter(
        const float* __restrict__ feats,   // (BN, C, H, W)
        const float* __restrict__ depth,   // (BN, H, W)
        const float* __restrict__ M,       // (N, 3, 4) composed transform
        float* __restrict__ out,           // (1, C, X, Y, Z)
        int npix) {
    const int p = blockIdx.x * blockDim.x + threadIdx.x;
    const int c = blockIdx.y;
    if (p >= npix) return;

    const int n  = p / (H_ * W_);
    const int hw = p - n * (H_ * W_);
    const int h  = hw / W_;
    const int w  = hw - h * W_;

    const float* fptr = feats + (((size_t)n * C_ + c) * H_ + h) * W_ + w;
    __builtin_prefetch(fptr, 0, 0);        // global_prefetch_b8: overlap with sqrt/matvec

    const float d    = depth[p];
    const float idxf = -0.5f + 0.5f * sqrtf(1.0f + 8.0f * (d - DMIN_) / BINF);
    if (!(idxf >= 0.0f && idxf < (float)D_)) return;   // also rejects NaN/Inf
    const int   di = (int)fminf(fmaxf(idxf, 0.0f), (float)(D_ - 1));
    const float ds = DMIN_ + BINF * (float)(di * (di + 1)) * 0.5f;

    const float px = (float)w * ds, py = (float)h * ds, pz = ds;
    const float* Mr = M + n * 12;
    const float wx = Mr[0] * px + Mr[1] * py + Mr[2]  * pz + Mr[3];
    const float wy = Mr[4] * px + Mr[5] * py + Mr[6]  * pz + Mr[7];
    const float wz = Mr[8] * px + Mr[9] * py + Mr[10] * pz + Mr[11];

    const int vx = (int)floorf(wx + 32.0f);
    const int vy = (int)floorf(wy + 32.0f);
    const int vz = (int)floorf(wz + 32.0f);
    if ((unsigned)vx >= (unsigned)X_ ||
        (unsigned)vy >= (unsigned)Y_ ||
        (unsigned)vz >= (unsigned)Z_) return;

    const size_t oidx = ((((size_t)c * X_ + vx) * Y_ + vy) * (size_t)Z_) + vz;
    atomicAdd(out + oidx, *fptr);          // global_atomic_add_f32, L2-resident cube
}

// ------------------------------------------------------------------
// Fallback fused scatter (no workspace dependency): recomputes camera
// math per thread. Used only if ws_size is too small for M.
// ------------------------------------------------------------------
__global__ __launch_bounds__(256) void lss_scatter_fused(
        const float* __restrict__ feats, const float* __restrict__ depth,
        const float* __restrict__ pose,  const float* __restrict__ intr,
        float* __restrict__ out, int npix) {
    const int p = blockIdx.x * blockDim.x + threadIdx.x;
    const int c = blockIdx.y;
    if (p >= npix) return;
    const int n  = p / (H_ * W_);
    const int hw = p - n * (H_ * W_);
    const int h  = hw / W_;
    const int w  = hw - h * W_;
    const float* fptr = feats + (((size_t)n * C_ + c) * H_ + h) * W_ + w;
    __builtin_prefetch(fptr, 0, 0);
    const float d    = depth[p];
    const float idxf = -0.5f + 0.5f * sqrtf(1.0f + 8.0f * (d - DMIN_) / BINF);
    if (!(idxf >= 0.0f && idxf < (float)D_)) return;
    const int   di = (int)fminf(fmaxf(idxf, 0.0f), (float)(D_ - 1));
    const float ds = DMIN_ + BINF * (float)(di * (di + 1)) * 0.5f;
    float inv[9];
    inv3x3(intr + n * 9, inv);
    const float px = (float)w * ds, py = (float)h * ds, pz = ds;
    const float cx = inv[0] * px + inv[1] * py + inv[2] * pz;
    const float cy = inv[3] * px + inv[4] * py + inv[5] * pz;
    const float cz = inv[6] * px + inv[7] * py + inv[8] * pz;
    const float* P = pose + n * 16;
    const float wx = P[0] * cx + P[1] * cy + P[2]  * cz + P[3];
    const float wy = P[4] * cx + P[5] * cy + P[6]  * cz + P[7];
    const float wz = P[8] * cx + P[9] * cy + P[10] * cz + P[11];
    const int vx = (int)floorf(wx + 32.0f);
    const int vy = (int)floorf(wy + 32.0f);
    const int vz = (int)floorf(wz + 32.0f);
    if ((unsigned)vx >= (unsigned)X_ ||
        (unsigned)vy >= (unsigned)Y_ ||
        (unsigned)vz >= (unsigned)Z_) return;
    atomicAdd(out + ((((size_t)c * X_ + vx) * Y_ + vy) * (size_t)Z_) + vz, *fptr);
}

extern "C" void kernel_launch(void* const* d_in, const int* in_sizes, int n_in,
                              void* d_out, int out_size, void* d_ws, size_t ws_size,
                              hipStream_t stream) {
    (void)n_in;
    const float* feats = (const float*)d_in[0];
    const float* depth = (const float*)d_in[1];
    const float* pose  = (const float*)d_in[2];
    const float* intr  = (const float*)d_in[3];
    float* out = (float*)d_out;

    // 1) zero the 134 MB cube (bandwidth floor of the whole op)
    const int n4   = out_size >> 2;
    const int tail = out_size - (n4 << 2);
    if (n4 > 0)  lss_zero4<<<dim3((n4 + 255) / 256), dim3(256), 0, stream>>>((float4*)out, n4);
    if (tail > 0) lss_zero1<<<dim3(1), dim3(256), 0, stream>>>(out, n4 << 2, out_size);

    const int npix = in_sizes[1];          // BN*H*W = 8400
    dim3 grid((npix + 255) / 256, C_);

    if (ws_size >= (size_t)(N_ * 12 * sizeof(float))) {
        // 2) precompose per-camera transforms (WMMA + scalar guard) into d_ws
        float* M = (float*)d_ws;
        lss_setup<<<dim3(1), dim3(32), 0, stream>>>(pose, intr, M);
        // 3) scatter-add
        lss_scatter<<<grid, dim3(256), 0, stream>>>(feats, depth, M, out, npix);
    } else {
        lss_scatter_fused<<<grid, dim3(256), 0, stream>>>(feats, depth, pose, intr, out, npix);
    }
}